// K_Sparse_Autoencoder_858993459789
// MI455X (gfx1250) — compile-verified
//
#include <hip/hip_runtime.h>
#include <hip/hip_bf16.h>

// ---------------------------------------------------------------------------
// Top-K Sparse Autoencoder, fused for MI455X (gfx1250, wave32, WMMA).
//
//   z1 = x @ W^T           [4096 x 16384], K=768   (103 GFLOP -> dominates)
//   a1 = topk32(z1)        (streaming per-row top-32, fused; z1 never in HBM)
//   z2 = a1 @ W + b_dec    (32 nnz/row sparse gather, L2-resident W)
//
// ROWS=32 per WG (128 WGs) halves the L2 W-stream vs ROWS=16 (6.4 GB total).
// 16 waves/WG, each wave owns a 16x32 tile; W chunks are register-prefetched
// so global L2 latency overlaps the v_wmma_f32_16x16x4_f32 compute phase.
// ---------------------------------------------------------------------------

typedef float v2f __attribute__((ext_vector_type(2)));
typedef float v8f __attribute__((ext_vector_type(8)));

#define BATCH     4096
#define DIM       768      // input dim (GEMM K)
#define FEAT      16384    // bottleneck (GEMM N)
#define TOPK      32

#define ROWS      32       // batch rows per workgroup (GEMM M tile)
#define FCH       256      // bottleneck columns per F-chunk (8 N-waves x 32)
#define KC        64       // K sub-chunk staged in LDS
#define NTHREADS  512      // 16 waves: 2 (M) x 8 (N)

#define XS        772      // LDS stride of x tile  (772%64==4 -> banks 4*lane; *4 mult of 16)
#define WS        68       // LDS stride of W tile  ( 68%64==4 -> banks 4*lane; *4 mult of 16)
#define ZS        260      // LDS stride of z chunk (260%64==4 -> scan banks 4*row)

__global__ __launch_bounds__(NTHREADS)
void sae_encode_topk(const float* __restrict__ x,
                     const float* __restrict__ W,
                     const float* __restrict__ b_enc,
                     float* __restrict__ outVals,
                     int*   __restrict__ outIdxs) {
  __shared__ float xT[ROWS * XS];     //  98,816 B : x tile [32 x 768]
  __shared__ float wT[FCH * WS];      //  69,632 B : W tile [256 n x 64 k]
  __shared__ float zC[ROWS * ZS];     //  33,280 B : z1 chunk [32 x 256]
  __shared__ float topV[ROWS * TOPK]; //   4,096 B
  __shared__ int   topI[ROWS * TOPK]; //   4,096 B    total ~205 KB -> 1 WG/WGP
  const int tid     = threadIdx.x;
  const int lane    = tid & 31;
  const int wave    = tid >> 5;        // 0..15
  const int mw      = wave >> 3;       // 0..1  (M half)
  const int nw      = wave & 7;        // 0..7  (N slab)
  const int hi      = lane >> 4;       // 0: K pair {0,1}, 1: K pair {2,3}
  const int l16     = lane & 15;
  const int rowBase = blockIdx.x * ROWS;

  // ---- stage x tile: 32 x 768 f32 = 6144 float4, 12 per thread -------------
  {
    const float* xg = x + (size_t)rowBase * DIM;
    #pragma unroll
    for (int i = 0; i < 12; ++i) {
      int lin = i * NTHREADS + tid;
      int r = lin / 192, c = lin % 192;             // c in float4 units
      float4 v = *(const float4*)(xg + (size_t)r * DIM + c * 4);
      *(float4*)&xT[r * XS + c * 4] = v;            // r*XS*4 = 3088r (16B aligned)
    }
  }
  for (int t = tid; t < ROWS * TOPK; t += NTHREADS) {
    topV[t] = -__builtin_inff();
    topI[t] = 0;
  }
  float thrMin = -__builtin_inff();   // per-row running min of top list (tid<32)
  int   thrPos = 0;

  const int nBase = nw * 32;          // this wave's 32-wide N slab within chunk
  const int mBase = mw * 16;          // this wave's 16-row M slab

  // W chunk staging indices: 4096 float4 per chunk, 8 per thread
  const int wN = tid >> 4;            // base n for this thread's first row... (see loop)
  (void)wN;

  for (int fc = 0; fc < FEAT / FCH; ++fc) {         // 64 F-chunks
    const int fBase = fc * FCH;
    v8f acc0 = {};
    v8f acc1 = {};

    // prefetch kc=0 of this F-chunk into registers
    float4 stage[8];
    {
      const float* wg = W + (size_t)fBase * DIM;
      #pragma unroll
      for (int i = 0; i < 8; ++i) {
        int lin = i * NTHREADS + tid;
        int n = lin >> 4, c = lin & 15;
        stage[i] = *(const float4*)(wg + (size_t)n * DIM + c * 4);
      }
    }

    for (int kc = 0; kc < DIM / KC; ++kc) {         // 12 K sub-chunks
      __syncthreads();                              // prior wT reads complete
      // ---- commit prefetched W chunk [256 n x 64 d] to LDS ----------------
      #pragma unroll
      for (int i = 0; i < 8; ++i) {
        int lin = i * NTHREADS + tid;
        int n = lin >> 4, c = lin & 15;
        *(float4*)&wT[n * WS + c * 4] = stage[i];   // n*WS*4 = 272n (16B aligned)
      }
      __syncthreads();

      // ---- issue prefetch of next chunk; overlaps the WMMA phase ----------
      if (kc + 1 < DIM / KC) {
        const float* wg = W + (size_t)fBase * DIM + (kc + 1) * KC;
        #pragma unroll
        for (int i = 0; i < 8; ++i) {
          int lin = i * NTHREADS + tid;
          int n = lin >> 4, c = lin & 15;
          stage[i] = *(const float4*)(wg + (size_t)n * DIM + c * 4);
        }
      }

      // ---- 16 WMMA K-steps of 4; A frag reused across two N-subtiles ------
      #pragma unroll
      for (int ks = 0; ks < 16; ++ks) {
        const int kk = ks * 4 + hi * 2;             // lane's K pair within chunk
        v2f a  = *(const v2f*)&xT[(mBase + l16) * XS + kc * KC + kk];
        v2f b0 = *(const v2f*)&wT[(nBase + l16) * WS + kk];
        v2f b1 = *(const v2f*)&wT[(nBase + 16 + l16) * WS + kk];
        acc0 = __builtin_amdgcn_wmma_f32_16x16x4_f32(
                   false, a, false, b0, (short)0, acc0, false, false);
        acc1 = __builtin_amdgcn_wmma_f32_16x16x4_f32(
                   false, a, false, b1, (short)0, acc1, false, false);
      }
    }

    // ---- spill z1 chunk to LDS (+ b_enc), C-layout: VGPR v -> M=v(+8) -----
    const float be0 = b_enc[fBase + nBase + l16];
    const float be1 = b_enc[fBase + nBase + 16 + l16];
    #pragma unroll
    for (int v = 0; v < 8; ++v) {
      const int m = mBase + v + hi * 8;
      zC[m * ZS + nBase + l16]      = acc0[v] + be0;
      zC[m * ZS + nBase + 16 + l16] = acc1[v] + be1;
    }
    __syncthreads();

    // ---- streaming top-32 update: lane r scans its row's 256 candidates ---
    if (tid < ROWS) {
      const int r = tid;
      for (int j = 0; j < FCH; ++j) {
        float vz = zC[r * ZS + j];
        if (vz > thrMin) {                          // rare after warm-up
          topV[r * TOPK + thrPos] = vz;
          topI[r * TOPK + thrPos] = fBase + j;
          float mn = topV[r * TOPK];
          int   mp = 0;
          #pragma unroll 4
          for (int s = 1; s < TOPK; ++s) {
            float tv = topV[r * TOPK + s];
            if (tv < mn) { mn = tv; mp = s; }
          }
          thrMin = mn; thrPos = mp;
        }
      }
    }
  }
  __syncthreads();

  // ---- emit (value, index) lists for the decode pass ----------------------
  for (int t = tid; t < ROWS * TOPK; t += NTHREADS) {
    int r = t >> 5, s = t & 31;
    outVals[(size_t)(rowBase + r) * TOPK + s] = topV[t];
    outIdxs[(size_t)(rowBase + r) * TOPK + s] = topI[t];
  }
}

// ---------------------------------------------------------------------------
// Decode: z2[b,:] = sum_t val[b,t] * W[idx[b,t],:] + b_dec.  W is L2-resident
// (50 MB < 192 MB L2); one block per batch row, 256 threads x 3 outputs.
// ---------------------------------------------------------------------------
__global__ __launch_bounds__(256)
void sae_decode(const float* __restrict__ vals,
                const int*   __restrict__ idxs,
                const float* __restrict__ W,
                const float* __restrict__ b_dec,
                float* __restrict__ out) {
  __shared__ float sv[TOPK];
  __shared__ int   si[TOPK];
  const int b   = blockIdx.x;
  const int tid = threadIdx.x;
  if (tid < TOPK) {
    sv[tid] = vals[(size_t)b * TOPK + tid];
    si[tid] = idxs[(size_t)b * TOPK + tid];
  }
  __syncthreads();

  float a0 = b_dec[tid];
  float a1 = b_dec[tid + 256];
  float a2 = b_dec[tid + 512];
  #pragma unroll 8
  for (int t = 0; t < TOPK; ++t) {
    const float v = sv[t];
    const float* wr = W + (size_t)si[t] * DIM;
    a0 += v * wr[tid];
    a1 += v * wr[tid + 256];
    a2 += v * wr[tid + 512];
  }
  float* o = out + (size_t)b * DIM;
  o[tid]       = a0;
  o[tid + 256] = a1;
  o[tid + 512] = a2;
}

extern "C" void kernel_launch(void* const* d_in, const int* in_sizes, int n_in,
                              void* d_out, int out_size, void* d_ws, size_t ws_size,
                              hipStream_t stream) {
  const float* x     = (const float*)d_in[0];   // [4096 x 768]
  const float* W     = (const float*)d_in[1];   // [16384 x 768]
  const float* b_enc = (const float*)d_in[2];   // [16384]
  const float* b_dec = (const float*)d_in[3];   // [768]
  // d_in[4] = k (device scalar); K=32 is baked in per the reference.

  float* wsVals = (float*)d_ws;                               // 4096*32 f32
  int*   wsIdxs = (int*)((char*)d_ws + (size_t)BATCH * TOPK * sizeof(float));
  float* out    = (float*)d_out;                              // [4096 x 768]

  sae_encode_topk<<<BATCH / ROWS, NTHREADS, 0, stream>>>(x, W, b_enc, wsVals, wsIdxs);
  sae_decode<<<BATCH, 256, 0, stream>>>(wsVals, wsIdxs, W, b_dec, out);
}